// TGCN_74380243632851
// MI455X (gfx1250) — compile-verified
//
#include <hip/hip_runtime.h>
#include <hip/hip_bf16.h>

typedef float v2f __attribute__((ext_vector_type(2)));
typedef float v8f __attribute__((ext_vector_type(8)));

#define WMMA4(A, B, C) \
  __builtin_amdgcn_wmma_f32_16x16x4_f32(false, (A), false, (B), (short)0, (C), false, false)

#define NN 50000
#define EE 800000
#define TT 8
#define FF 128
#define HH 128
#define CC 10
#define GG 64

__device__ __forceinline__ float sigmoid_f(float x) { return 1.0f / (1.0f + __expf(-x)); }
__device__ __forceinline__ float tanh_f(float x)    { return 2.0f / (1.0f + __expf(-2.0f * x)) - 1.0f; }

// ---------------------------------------------------------------- utilities
__global__ void fill_kernel(float* __restrict__ p, float v, int n) {
  int i = blockIdx.x * blockDim.x + threadIdx.x;
  if (i < n) p[i] = v;
}

__global__ void deg_kernel(const int* __restrict__ dst, float* __restrict__ deg, int e) {
  int i = blockIdx.x * blockDim.x + threadIdx.x;
  if (i < e) unsafeAtomicAdd(&deg[dst[i]], 1.0f);
}

__global__ void rsqrt_kernel(float* __restrict__ d, int n) {
  int i = blockIdx.x * blockDim.x + threadIdx.x;
  if (i < n) d[i] = rsqrtf(d[i]);
}

__global__ void transpose128_kernel(const float* __restrict__ W, float* __restrict__ WT) {
  int i = blockIdx.x * blockDim.x + threadIdx.x;  // 16384 threads
  int k = i >> 7, j = i & 127;                    // W[k*128 + j]
  WT[j * 128 + k] = W[i];
}

// ------------------------------------------------ GCN feature GEMM (WMMA f32)
// xw = x_t @ W_gcn  (via WT[j,k] for contiguous B loads); also agg = xw * dis^2
__global__ __launch_bounds__(32) void gemm_xw_kernel(
    const float* __restrict__ xt,   // [N,128] timestep slice
    const float* __restrict__ wT,   // [128,128] = W_gcn^T : wT[j*128+k]
    const float* __restrict__ dis,  // [N]
    float* __restrict__ xw,         // [N,128]
    float* __restrict__ agg)        // [N,128]
{
  __shared__ float xa[16 * 128];
  const int r0 = blockIdx.x * 16;      // 3125 blocks, N % 16 == 0
  const int lane = threadIdx.x;        // one wave32 per block

  // stage 16x128 A rows into LDS (float4)
  const float4* s4 = (const float4*)(xt + (size_t)r0 * 128);
  float4* d4 = (float4*)xa;
#pragma unroll
  for (int i = 0; i < 16; ++i) d4[lane + i * 32] = s4[lane + i * 32];
  __syncthreads();

  const int mm = lane & 15;
  const int kh = (lane >> 4) << 1;  // 0 or 2

  // hoist self-norm factors (dis^2) for this wave's 16 rows
  float sn[8];
#pragma unroll
  for (int d = 0; d < 8; ++d) {
    float dv = dis[r0 + d + ((lane >> 4) << 3)];
    sn[d] = dv * dv;
  }

  for (int c = 0; c < 8; ++c) {
    v8f acc0 = {}, acc1 = {};  // two independent WMMA dependency chains
    const float* ab = xa + mm * 128 + kh;
    const float* bb = wT + (c * 16 + mm) * 128 + kh;
#pragma unroll
    for (int kc = 0; kc < 32; kc += 2) {
      v2f a0 = *(const v2f*)(ab + kc * 4);
      v2f b0 = *(const v2f*)(bb + kc * 4);
      v2f a1 = *(const v2f*)(ab + kc * 4 + 4);
      v2f b1 = *(const v2f*)(bb + kc * 4 + 4);
      acc0 = WMMA4(a0, b0, acc0);
      acc1 = WMMA4(a1, b1, acc1);
    }
    v8f acc = acc0 + acc1;
#pragma unroll
    for (int d = 0; d < 8; ++d) {
      int m = d + ((lane >> 4) << 3);
      size_t idx = (size_t)(r0 + m) * 128 + c * 16 + mm;
      float v = acc[d];
      xw[idx] = v;
      agg[idx] = v * sn[d];  // self-loop term D^-1 * xw
    }
  }
}

// ------------------------------------------------------------ edge scatter
// agg[dst] += xw[src] * dis[src]*dis[dst]   (wave per edge, coalesced)
__global__ __launch_bounds__(256) void scatter_kernel(
    const int* __restrict__ esrc, const int* __restrict__ edst,
    const float* __restrict__ dis,
    const float* __restrict__ xw, float* __restrict__ agg)
{
  int e = blockIdx.x * 8 + (threadIdx.x >> 5);  // E % 8 == 0
  int lane = threadIdx.x & 31;
  int s = esrc[e], d = edst[e];
  float norm = dis[s] * dis[d];
  const float* xr = xw + (size_t)s * 128;
  float* ar = agg + (size_t)d * 128;
#pragma unroll
  for (int f = 0; f < 4; ++f) {
    int k = lane + f * 32;
    unsafeAtomicAdd(ar + k, xr[k] * norm);
  }
}

// --------------------------------------------------------------- GRU step
// x = relu(agg + b_gcn); gx = x@W_ih^T + b_ih; gh = h@W_hh^T + b_hh
// r=sig(xr+hr); z=sig(xz+hz); n=tanh(xn + r*hn); h = (1-z)*n + z*h   (in place)
__global__ __launch_bounds__(32) void gru_step_kernel(
    const float* __restrict__ agg,   // [N,128] pre-activation, no bias
    const float* __restrict__ b_gcn, // [128]
    const float* __restrict__ W_ih,  // [384,128]
    const float* __restrict__ W_hh,  // [384,128]
    const float* __restrict__ b_ih,  // [384]
    const float* __restrict__ b_hh,  // [384]
    float* __restrict__ h)           // [N,128] in-out
{
  __shared__ float xs[16 * 128];
  __shared__ float hs[16 * 128];
  const int r0 = blockIdx.x * 16;
  const int lane = threadIdx.x;

  // stage relu(agg + b_gcn) and h rows into LDS
  const float4* ap = (const float4*)(agg + (size_t)r0 * 128);
  const float4* hp = (const float4*)(h + (size_t)r0 * 128);
  const float4* bg = (const float4*)b_gcn;
  float4* xs4 = (float4*)xs;
  float4* hs4 = (float4*)hs;
#pragma unroll
  for (int i = 0; i < 16; ++i) {
    int idx = lane + i * 32;  // float4 index, 0..511
    float4 a = ap[idx];
    float4 b = bg[idx & 31];
    float4 r;
    r.x = fmaxf(a.x + b.x, 0.0f);
    r.y = fmaxf(a.y + b.y, 0.0f);
    r.z = fmaxf(a.z + b.z, 0.0f);
    r.w = fmaxf(a.w + b.w, 0.0f);
    xs4[idx] = r;
    hs4[idx] = hp[idx];
  }
  __syncthreads();

  const int mm = lane & 15;
  const int kh = (lane >> 4) << 1;

  for (int c = 0; c < 8; ++c) {
    v8f axr = {}, axz = {}, axn = {}, ahr = {}, ahz = {}, ahn = {};
    const int col = c * 16 + mm;
    const float* ax = xs + mm * 128 + kh;
    const float* ah = hs + mm * 128 + kh;
    const float* wir = W_ih + (size_t)(0 * 128 + col) * 128 + kh;
    const float* wiz = W_ih + (size_t)(1 * 128 + col) * 128 + kh;
    const float* win = W_ih + (size_t)(2 * 128 + col) * 128 + kh;
    const float* whr = W_hh + (size_t)(0 * 128 + col) * 128 + kh;
    const float* whz = W_hh + (size_t)(1 * 128 + col) * 128 + kh;
    const float* whn = W_hh + (size_t)(2 * 128 + col) * 128 + kh;
#pragma unroll
    for (int kc = 0; kc < 32; ++kc) {
      v2f aX = *(const v2f*)(ax + kc * 4);
      v2f aH = *(const v2f*)(ah + kc * 4);
      v2f br = *(const v2f*)(wir + kc * 4);
      v2f bz = *(const v2f*)(wiz + kc * 4);
      v2f bn = *(const v2f*)(win + kc * 4);
      v2f cr = *(const v2f*)(whr + kc * 4);
      v2f cz = *(const v2f*)(whz + kc * 4);
      v2f cn = *(const v2f*)(whn + kc * 4);
      axr = WMMA4(aX, br, axr);
      axz = WMMA4(aX, bz, axz);
      axn = WMMA4(aX, bn, axn);
      ahr = WMMA4(aH, cr, ahr);
      ahz = WMMA4(aH, cz, ahz);
      ahn = WMMA4(aH, cn, ahn);
    }
    const float bxr = b_ih[col], bxz = b_ih[128 + col], bxn = b_ih[256 + col];
    const float bhr = b_hh[col], bhz = b_hh[128 + col], bhn = b_hh[256 + col];
#pragma unroll
    for (int d = 0; d < 8; ++d) {
      int m = d + ((lane >> 4) << 3);
      float hold = hs[m * 128 + col];
      float r = sigmoid_f(axr[d] + ahr[d] + bxr + bhr);
      float z = sigmoid_f(axz[d] + ahz[d] + bxz + bhz);
      float n = tanh_f(axn[d] + bxn + r * (ahn[d] + bhn));
      h[(size_t)(r0 + m) * 128 + col] = (1.0f - z) * n + z * hold;
    }
  }
}

// ------------------------------------------------------ FC + graph mean pool
__global__ __launch_bounds__(256) void pool_kernel(
    const float* __restrict__ h, const float* __restrict__ W_fc,
    const float* __restrict__ b_fc, const int* __restrict__ batch,
    float* __restrict__ sums, float* __restrict__ cnt, int n)
{
  int i = blockIdx.x * blockDim.x + threadIdx.x;
  if (i >= n) return;
  float acc[CC];
#pragma unroll
  for (int c = 0; c < CC; ++c) acc[c] = b_fc[c];
  const float* hr = h + (size_t)i * 128;
  for (int k = 0; k < 128; ++k) {
    float hv = hr[k];
#pragma unroll
    for (int c = 0; c < CC; ++c) acc[c] += hv * W_fc[k * CC + c];
  }
  int g = batch[i];
#pragma unroll
  for (int c = 0; c < CC; ++c) unsafeAtomicAdd(&sums[g * CC + c], acc[c]);
  unsafeAtomicAdd(&cnt[g], 1.0f);
}

__global__ void finalize_kernel(const float* __restrict__ sums,
                                const float* __restrict__ cnt,
                                float* __restrict__ out) {
  int i = threadIdx.x;  // 640
  if (i < GG * CC) out[i] = sums[i] / fmaxf(cnt[i / CC], 1.0f);
}

// ------------------------------------------------------------------- driver
extern "C" void kernel_launch(void* const* d_in, const int* in_sizes, int n_in,
                              void* d_out, int out_size, void* d_ws, size_t ws_size,
                              hipStream_t stream) {
  const float* x_seq = (const float*)d_in[0];
  const int*   edge  = (const int*)d_in[1];
  const int*   batch = (const int*)d_in[2];
  const float* W_gcn = (const float*)d_in[3];
  const float* b_gcn = (const float*)d_in[4];
  const float* W_ih  = (const float*)d_in[5];
  const float* W_hh  = (const float*)d_in[6];
  const float* b_ih  = (const float*)d_in[7];
  const float* b_hh  = (const float*)d_in[8];
  const float* W_fc  = (const float*)d_in[9];
  const float* b_fc  = (const float*)d_in[10];
  const int* esrc = edge;
  const int* edst = edge + EE;

  float* ws = (float*)d_ws;
  size_t o = 0;
  float* dis  = ws + o; o += 50048;               // deg -> rsqrt in place
  float* wT   = ws + o; o += 16384;               // W_gcn^T
  float* xw   = ws + o; o += (size_t)NN * HH;     // per-t xw
  float* agg  = ws + o; o += (size_t)NN * HH;     // per-t aggregate
  float* h    = ws + o; o += (size_t)NN * HH;     // GRU hidden (in place)
  float* sums = ws + o; o += GG * CC;
  float* cnt  = ws + o; o += GG;

  const int B = 256;
  fill_kernel<<<(NN + B - 1) / B, B, 0, stream>>>(dis, 1.0f, NN);
  fill_kernel<<<((NN * HH) + B - 1) / B, B, 0, stream>>>(h, 0.0f, NN * HH);
  fill_kernel<<<(GG * CC + GG + B - 1) / B, B, 0, stream>>>(sums, 0.0f, GG * CC + GG);
  deg_kernel<<<(EE + B - 1) / B, B, 0, stream>>>(edst, dis, EE);
  rsqrt_kernel<<<(NN + B - 1) / B, B, 0, stream>>>(dis, NN);
  transpose128_kernel<<<64, 256, 0, stream>>>(W_gcn, wT);

  for (int t = 0; t < TT; ++t) {
    const float* xt = x_seq + (size_t)t * NN * FF;
    gemm_xw_kernel<<<NN / 16, 32, 0, stream>>>(xt, wT, dis, xw, agg);
    scatter_kernel<<<EE / 8, 256, 0, stream>>>(esrc, edst, dis, xw, agg);
    gru_step_kernel<<<NN / 16, 32, 0, stream>>>(agg, b_gcn, W_ih, W_hh, b_ih, b_hh, h);
  }

  pool_kernel<<<(NN + B - 1) / B, B, 0, stream>>>(h, W_fc, b_fc, batch, sums, cnt, NN);
  finalize_kernel<<<1, 640, 0, stream>>>(sums, cnt, (float*)d_out);
}